// GNNTaskEncoder_66288525246861
// MI455X (gfx1250) — compile-verified
//
#include <hip/hip_runtime.h>
#include <math.h>

#define HIDDEN    256
#define HEADS     4
#define HEAD_DIM  64
#define EDGE_DIM  3
#define NUM_GRAPHS 64
#define LAYERS    3
#define NEG_SLOPE 0.2f

// GEMM tiling
#define MT        64    // output rows per block
#define KC        32    // K-slice staged in LDS
#define AS_STRIDE 36    // 32 + 4 pad (float4-aligned)
#define BS_STRIDE 260   // 256 + 4 pad

typedef __attribute__((ext_vector_type(2))) float v2f;
typedef __attribute__((ext_vector_type(8))) float v8f;

static __device__ __forceinline__ float elu1(float x) {
  return x > 0.f ? x : (__expf(x) - 1.f);
}

// ---------------------------------------------------------------------------
// WMMA f32 GEMM:  C[M,N] = A[M,K] @ B[K,N] + bias[N]   (N must be 256)
// 256-thread block = 8 waves -> 64x256 output tile.
// Wave w: col-strip (w&3)*64, row-pair (w>>2)*32 (two 16-row tiles, acc[2][4]).
// Per K-step: 2 A-frag b64 loads + 4 B-frag loads feed 8 V_WMMA_F32_16X16X4.
// K staged in 32-wide LDS slices (A: 9.2KB, B: 33.3KB, padded rows).
// ---------------------------------------------------------------------------
__global__ __launch_bounds__(256) void gemm_bias_wmma(
    const float* __restrict__ A, const float* __restrict__ B,
    const float* __restrict__ bias, float* __restrict__ C,
    int M, int K, int N) {
  __shared__ float As[MT * AS_STRIDE];
  __shared__ float Bs[KC * BS_STRIDE];

  const int tid   = threadIdx.x;
  const int lane  = tid & 31;
  const int wave  = tid >> 5;
  const int row   = lane & 15;    // also column-within-16-tile for B/C frags
  const int khalf = lane >> 4;    // 0 or 1
  const int kh2   = khalf << 1;
  const int sw    = (wave & 3) << 6;   // col strip 0/64/128/192
  const int mbase = (wave >> 2) << 5;  // row base within tile: 0 or 32
  const int m0    = blockIdx.x << 6;

  v8f acc[2][4] = {};

#define GEMM_STEP(k4)                                                          \
  {                                                                            \
    const v2f a0 = *(const v2f*)&As[(mbase + row) * AS_STRIDE + (k4) + kh2];   \
    const v2f a1 = *(const v2f*)&As[(mbase + 16 + row) * AS_STRIDE + (k4) + kh2]; \
    const float* b0 = &Bs[(size_t)((k4) + kh2) * BS_STRIDE + sw + row];        \
    _Pragma("unroll")                                                          \
    for (int t = 0; t < 4; ++t) {                                              \
      v2f b;                                                                   \
      b.x = b0[t * 16];                                                        \
      b.y = b0[t * 16 + BS_STRIDE];                                            \
      acc[0][t] = __builtin_amdgcn_wmma_f32_16x16x4_f32(                       \
          false, a0, false, b, (short)0, acc[0][t], false, false);             \
      acc[1][t] = __builtin_amdgcn_wmma_f32_16x16x4_f32(                       \
          false, a1, false, b, (short)0, acc[1][t], false, false);             \
    }                                                                          \
  }

  for (int kk = 0; kk < K; kk += KC) {
    const int kc = (K - kk) < KC ? (K - kk) : KC;

    // prefetch next A K-slice into caches (global_prefetch_b8)
    if (kk + KC < K) {
      int pr = m0 + (tid & 63);
      if (pr >= M) pr = M - 1;
      __builtin_prefetch(&A[(size_t)pr * K + kk + KC], 0, 0);
    }

    // stage A tile: 64 rows x kc cols (512 float4, 2 per thread)
#pragma unroll
    for (int j = 0; j < 2; ++j) {
      const int idx = j * 256 + tid;  // 0..511
      const int r = idx >> 3;         // 0..63
      const int c = (idx & 7) << 2;   // 0..28
      float4 v = make_float4(0.f, 0.f, 0.f, 0.f);
      if (c < kc && (m0 + r) < M)
        v = *(const float4*)&A[(size_t)(m0 + r) * K + kk + c];
      *(float4*)&As[r * AS_STRIDE + c] = v;
    }
    // stage B tile: kc rows x 256 cols (2048 float4, 8 per thread)
#pragma unroll
    for (int j = 0; j < 8; ++j) {
      const int idx = j * 256 + tid;
      const int r = idx >> 6;         // 0..31
      const int c = (idx & 63) << 2;  // 0..252
      if (r < kc)
        *(float4*)&Bs[r * BS_STRIDE + c] =
            *(const float4*)&B[(size_t)(kk + r) * N + c];
    }
    __syncthreads();

    if (kc == KC) {
      // hot path: branch-free fully unrolled 8 K-steps
#pragma unroll
      for (int k4 = 0; k4 < KC; k4 += 4) GEMM_STEP(k4);
    } else {
      for (int k4 = 0; k4 < kc; k4 += 4) GEMM_STEP(k4);
    }
    __syncthreads();
  }
#undef GEMM_STEP

#pragma unroll
  for (int mt = 0; mt < 2; ++mt) {
    const int rbase = m0 + mbase + mt * 16 + (khalf << 3);
#pragma unroll
    for (int t = 0; t < 4; ++t) {
      const int col = sw + t * 16 + row;
      const float bv = bias[col];
#pragma unroll
      for (int i = 0; i < 8; ++i) {
        const int r = rbase + i;
        if (r < M) C[(size_t)r * N + col] = acc[mt][t][i] + bv;
      }
    }
  }
}

// ---------------------------------------------------------------------------
// Fused LayerNorm + ELU over rows of width HIDDEN. One 256-thread block/row.
// ---------------------------------------------------------------------------
__global__ void ln_elu_kernel(const float* __restrict__ in,
                              const float* __restrict__ g,
                              const float* __restrict__ b,
                              float* __restrict__ out) {
  __shared__ float red[HIDDEN];
  const int rowb = blockIdx.x;
  const int tid  = threadIdx.x;
  float v = in[(size_t)rowb * HIDDEN + tid];
  red[tid] = v;
  __syncthreads();
  for (int s = HIDDEN / 2; s > 0; s >>= 1) {
    if (tid < s) red[tid] += red[tid + s];
    __syncthreads();
  }
  float mu = red[0] * (1.f / HIDDEN);
  __syncthreads();
  float d = v - mu;
  red[tid] = d * d;
  __syncthreads();
  for (int s = HIDDEN / 2; s > 0; s >>= 1) {
    if (tid < s) red[tid] += red[tid + s];
    __syncthreads();
  }
  float var = red[0] * (1.f / HIDDEN);
  float y = d * rsqrtf(var + 1e-5f) * g[tid] + b[tid];
  out[(size_t)rowb * HIDDEN + tid] = elu1(y);
}

// Per-node attention dot products: sal/dal[n,h] = hp[n,h,:] . a_{s,d}[h,:]
__global__ void node_alpha_kernel(const float* __restrict__ hp,
                                  const float* __restrict__ a_s,
                                  const float* __restrict__ a_d,
                                  float* __restrict__ sal,
                                  float* __restrict__ dal, int Nn) {
  int t = blockIdx.x * blockDim.x + threadIdx.x;
  if (t >= Nn * HEADS) return;
  int n = t >> 2, hh = t & 3;
  const float* hv = hp + (size_t)n * HIDDEN + hh * HEAD_DIM;
  const float* as = a_s + hh * HEAD_DIM;
  const float* ad = a_d + hh * HEAD_DIM;
  float ss = 0.f, sd = 0.f;
#pragma unroll 8
  for (int d = 0; d < HEAD_DIM; ++d) { ss += hv[d] * as[d]; sd += hv[d] * ad[d]; }
  sal[t] = ss; dal[t] = sd;
}

// ce[k,h] = sum_d We[k, h*64+d] * a_e[h,d]  -> collapses edge projection
__global__ void cedge_kernel(const float* __restrict__ We,
                             const float* __restrict__ a_e,
                             float* __restrict__ ce) {
  int t = threadIdx.x;
  if (t >= EDGE_DIM * HEADS) return;
  int k = t >> 2, hh = t & 3;
  float s = 0.f;
  for (int d = 0; d < HEAD_DIM; ++d)
    s += We[(size_t)k * HIDDEN + hh * HEAD_DIM + d] * a_e[hh * HEAD_DIM + d];
  ce[k * HEADS + hh] = s;
}

__global__ void fill_kernel(float* __restrict__ p, float v, size_t n) {
  size_t i = (size_t)blockIdx.x * blockDim.x + threadIdx.x;
  if (i < n) p[i] = v;
}

static __device__ __forceinline__ unsigned int f2key(float f) {
  unsigned int bits = __float_as_uint(f);
  return (bits & 0x80000000u) ? ~bits : (bits | 0x80000000u);
}

// Pass 1: one thread per edge, all 4 heads. float4 sal/dal loads,
// leaky-relu logits, segment max via monotonic-key u32 atomicMax.
__global__ void edge_logits_kernel(const int* __restrict__ ei,
                                   const float* __restrict__ ea,
                                   const float* __restrict__ sal,
                                   const float* __restrict__ dal,
                                   const float* __restrict__ ce,
                                   float* __restrict__ lg,
                                   unsigned int* __restrict__ mkey, int E) {
  int e = blockIdx.x * blockDim.x + threadIdx.x;
  if (e >= E) return;
  const int src = ei[e], dst = ei[E + e];
  const float4 s4 = *(const float4*)&sal[src * HEADS];
  const float4 d4 = *(const float4*)&dal[dst * HEADS];
  const float e0 = ea[e * 3 + 0], e1 = ea[e * 3 + 1], e2 = ea[e * 3 + 2];
  const float* sp = (const float*)&s4;
  const float* dp = (const float*)&d4;
  float lv[HEADS];
#pragma unroll
  for (int hh = 0; hh < HEADS; ++hh) {
    float l = sp[hh] + dp[hh] + e0 * ce[hh] + e1 * ce[HEADS + hh] + e2 * ce[2 * HEADS + hh];
    l = l > 0.f ? l : NEG_SLOPE * l;
    lv[hh] = l;
    atomicMax(&mkey[dst * HEADS + hh], f2key(l));
  }
  *(float4*)&lg[(size_t)e * HEADS] = make_float4(lv[0], lv[1], lv[2], lv[3]);
}

// Decode keys -> floats; non-finite (no incoming edges) -> 0 per reference
__global__ void decode_m_kernel(const unsigned int* __restrict__ mkey,
                                float* __restrict__ mfl, int n) {
  int t = blockIdx.x * blockDim.x + threadIdx.x;
  if (t >= n) return;
  unsigned int key = mkey[t];
  unsigned int bits = (key & 0x80000000u) ? (key & 0x7FFFFFFFu) : ~key;
  float f = __uint_as_float(bits);
  if (!(f > -3.0e38f && f < 3.0e38f)) f = 0.f;  // NaN/inf -> 0
  mfl[t] = f;
}

// Pass 2: one thread per edge: ex = exp(logit - m[dst]) in place, denom += ex
__global__ void edge_exp_kernel(const int* __restrict__ ei,
                                float* __restrict__ lg,
                                const float* __restrict__ mfl,
                                float* __restrict__ den, int E) {
  int e = blockIdx.x * blockDim.x + threadIdx.x;
  if (e >= E) return;
  const int dst = ei[E + e];
  const float4 l4 = *(const float4*)&lg[(size_t)e * HEADS];
  const float4 m4 = *(const float4*)&mfl[dst * HEADS];
  const float* lp = (const float*)&l4;
  const float* mp = (const float*)&m4;
  float ev[HEADS];
#pragma unroll
  for (int hh = 0; hh < HEADS; ++hh) {
    ev[hh] = __expf(lp[hh] - mp[hh]);
    atomicAdd(&den[dst * HEADS + hh], ev[hh]);
  }
  *(float4*)&lg[(size_t)e * HEADS] = make_float4(ev[0], ev[1], ev[2], ev[3]);
}

// Pass 3: message scatter. 64 threads/edge, float4 gather + 4 atomic adds.
__global__ void edge_message_kernel(const int* __restrict__ ei,
                                    const float* __restrict__ lg,
                                    const float* __restrict__ den,
                                    const float* __restrict__ hp,
                                    float* __restrict__ aggr, int E) {
  long long gt = (long long)blockIdx.x * blockDim.x + threadIdx.x;
  if (gt >= (long long)E * 64) return;
  int e = (int)(gt >> 6);
  int i = (int)(gt & 63);
  int src = ei[e], dst = ei[E + e];
  int hh = i >> 4;
  float d = fmaxf(den[dst * HEADS + hh], 1e-16f);
  float alpha = lg[(size_t)e * HEADS + hh] / d;
  const float4 hv = *(const float4*)(hp + (size_t)src * HIDDEN + i * 4);
  float* out = aggr + (size_t)dst * HIDDEN + i * 4;
  atomicAdd(out + 0, hv.x * alpha);
  atomicAdd(out + 1, hv.y * alpha);
  atomicAdd(out + 2, hv.z * alpha);
  atomicAdd(out + 3, hv.w * alpha);
}

__global__ void residual_elu_kernel(const float* __restrict__ aggr,
                                    float* __restrict__ h, size_t n) {
  size_t t = (size_t)blockIdx.x * blockDim.x + threadIdx.x;
  if (t >= n) return;
  float v = aggr[t] + h[t];
  h[t] = elu1(v);
}

__global__ void pool_accum_kernel(const float* __restrict__ emb,
                                  const int* __restrict__ batch,
                                  float* __restrict__ sums,
                                  float* __restrict__ cnt, int Nn) {
  long long t = (long long)blockIdx.x * blockDim.x + threadIdx.x;
  if (t >= (long long)Nn * HIDDEN) return;
  int n = (int)(t >> 8);
  int c = (int)(t & 255);
  int g = batch[n];
  atomicAdd(&sums[(size_t)g * HIDDEN + c], emb[t]);
  if (c == 0) atomicAdd(&cnt[g], 1.f);
}

__global__ void pool_div_kernel(const float* __restrict__ sums,
                                const float* __restrict__ cnt,
                                float* __restrict__ out) {
  int t = blockIdx.x * blockDim.x + threadIdx.x;
  out[t] = sums[t] / fmaxf(cnt[t >> 8], 1.f);
}

static inline int cdiv(long long a, int b) { return (int)((a + b - 1) / b); }

extern "C" void kernel_launch(void* const* d_in, const int* in_sizes, int n_in,
                              void* d_out, int out_size, void* d_ws, size_t ws_size,
                              hipStream_t stream) {
  const float* x        = (const float*)d_in[0];
  const int*   ei       = (const int*)d_in[1];
  const float* ea       = (const float*)d_in[2];
  const int*   batch    = (const int*)d_in[3];
  const float* W_in     = (const float*)d_in[4];
  const float* b_in     = (const float*)d_in[5];
  const float* ln_in_g  = (const float*)d_in[6];
  const float* ln_in_b  = (const float*)d_in[7];
  const float* W_gat    = (const float*)d_in[8];
  const float* b_gat    = (const float*)d_in[9];
  const float* W_e      = (const float*)d_in[10];
  const float* a_src    = (const float*)d_in[11];
  const float* a_dst    = (const float*)d_in[12];
  const float* a_edge   = (const float*)d_in[13];
  const float* W_out    = (const float*)d_in[14];
  const float* b_out    = (const float*)d_in[15];
  const float* ln_out_g = (const float*)d_in[16];
  const float* ln_out_b = (const float*)d_in[17];

  const int Nn = in_sizes[0] / 8;   // 50000
  const int E  = in_sizes[1] / 2;   // 800000

  float* ws = (float*)d_ws;
  float* h    = ws;  ws += (size_t)Nn * HIDDEN;
  float* hp   = ws;  ws += (size_t)Nn * HIDDEN;
  float* aggr = ws;  ws += (size_t)Nn * HIDDEN;
  float* sal  = ws;  ws += (size_t)Nn * HEADS;
  float* dal  = ws;  ws += (size_t)Nn * HEADS;
  unsigned int* mkey = (unsigned int*)ws; ws += (size_t)Nn * HEADS;
  float* mfl  = ws;  ws += (size_t)Nn * HEADS;
  float* den  = ws;  ws += (size_t)Nn * HEADS;
  float* lg   = ws;  ws += (size_t)E * HEADS;
  float* ce   = ws;  ws += 16;
  float* psum = ws;  ws += (size_t)NUM_GRAPHS * HIDDEN;
  float* pcnt = ws;  ws += NUM_GRAPHS;
  (void)ws_size; (void)n_in; (void)out_size;

  float* graph_emb = (float*)d_out;
  float* node_emb  = (float*)d_out + (size_t)NUM_GRAPHS * HIDDEN;

  const dim3 gB(256);
  const dim3 gG(cdiv(Nn, MT));

  // input projection + LN + ELU
  gemm_bias_wmma<<<gG, gB, 0, stream>>>(x, W_in, b_in, hp, Nn, 8, HIDDEN);
  ln_elu_kernel<<<Nn, HIDDEN, 0, stream>>>(hp, ln_in_g, ln_in_b, h);

  for (int l = 0; l < LAYERS; ++l) {
    gemm_bias_wmma<<<gG, gB, 0, stream>>>(h, W_gat + (size_t)l * HIDDEN * HIDDEN,
                                          b_gat + l * HIDDEN, hp, Nn, HIDDEN, HIDDEN);
    node_alpha_kernel<<<cdiv((long long)Nn * HEADS, 256), 256, 0, stream>>>(
        hp, a_src + l * HEADS * HEAD_DIM, a_dst + l * HEADS * HEAD_DIM, sal, dal, Nn);
    cedge_kernel<<<1, 16, 0, stream>>>(W_e + (size_t)l * EDGE_DIM * HIDDEN,
                                       a_edge + l * HEADS * HEAD_DIM, ce);
    fill_kernel<<<cdiv((long long)Nn * HEADS, 256), 256, 0, stream>>>((float*)mkey, 0.f, (size_t)Nn * HEADS);
    fill_kernel<<<cdiv((long long)Nn * HEADS, 256), 256, 0, stream>>>(den, 0.f, (size_t)Nn * HEADS);
    fill_kernel<<<cdiv((long long)Nn * HIDDEN, 256), 256, 0, stream>>>(aggr, 0.f, (size_t)Nn * HIDDEN);

    edge_logits_kernel<<<cdiv(E, 256), 256, 0, stream>>>(ei, ea, sal, dal, ce, lg, mkey, E);
    decode_m_kernel<<<cdiv((long long)Nn * HEADS, 256), 256, 0, stream>>>(mkey, mfl, Nn * HEADS);
    edge_exp_kernel<<<cdiv(E, 256), 256, 0, stream>>>(ei, lg, mfl, den, E);
    edge_message_kernel<<<cdiv((long long)E * 64, 256), 256, 0, stream>>>(ei, lg, den, hp, aggr, E);
    residual_elu_kernel<<<cdiv((long long)Nn * HIDDEN, 256), 256, 0, stream>>>(aggr, h, (size_t)Nn * HIDDEN);
  }

  // output projection + LN + ELU -> node_emb
  gemm_bias_wmma<<<gG, gB, 0, stream>>>(h, W_out, b_out, hp, Nn, HIDDEN, HIDDEN);
  ln_elu_kernel<<<Nn, HIDDEN, 0, stream>>>(hp, ln_out_g, ln_out_b, node_emb);

  // global mean pool per graph
  fill_kernel<<<cdiv(NUM_GRAPHS * HIDDEN, 256), 256, 0, stream>>>(psum, 0.f, (size_t)NUM_GRAPHS * HIDDEN);
  fill_kernel<<<1, 256, 0, stream>>>(pcnt, 0.f, (size_t)NUM_GRAPHS);
  pool_accum_kernel<<<cdiv((long long)Nn * HIDDEN, 256), 256, 0, stream>>>(node_emb, batch, psum, pcnt, Nn);
  pool_div_kernel<<<NUM_GRAPHS, HIDDEN, 0, stream>>>(psum, pcnt, graph_emb);
}